// LlamaDecodeAttention_82489141887104
// MI455X (gfx1250) — compile-verified
//
#include <hip/hip_runtime.h>

// ---------------------------------------------------------------------------
// LlamaDecodeAttention for MI455X (gfx1250, wave32).
//
// Sizes: BATCH=16, DIM=4096, N_HEADS=32, HEAD_DIM=128, CACHE_LEN=4096 (fp32).
//
// Roofline: traffic = wq(64MB) + wv(64MB) + wo(64MB) + K/V cache (2GB)
//           ~= 2.25 GB  ->  ~97us @ 23.3 TB/s.  FLOPs ~2 GF -> memory bound.
// Algebra:  reference applies a *separate* softmax to the single new token,
//           so its probability is exactly 1 -> new_out == v_new and wk/k_new
//           never influence the output. We skip the wk GEMM entirely (-64MB).
// ---------------------------------------------------------------------------

typedef float v2f __attribute__((ext_vector_type(2)));
typedef float v4f __attribute__((ext_vector_type(4)));
typedef float v8f __attribute__((ext_vector_type(8)));

#define DIM       4096
#define NHEADS    32
#define HDIM      128
#define CACHE_LEN 4096
#define BATCH     16

// ---------------------------------------------------------------------------
// GEMM  y[b,o] = sum_d x[b,d] * w[o,d]      (y = x @ w.T)
//   x: (16, 4096)   w: (4096, 4096)   y: (16, 4096)
//
// M = 16 (all batches), one 16-wide N tile per workgroup (grid.x = 256),
// split-K: 8 waves x 512, each wave runs V_WMMA_F32_16X16X4_F32 over its
// K-chunk, partial 16x16 C tiles reduced through LDS.
//
// Fragment layouts per CDNA5 ISA 7.12.2 (32-bit A 16x4 / mirrored B 4x16):
//   lane L: m|n = L%16 ;  VGPR j holds K = j + 2*(L/16)  -> contiguous float2.
// C/D: VGPR r, lane L -> M = r + 8*(L/16), N = L%16.
// ---------------------------------------------------------------------------
__global__ __launch_bounds__(256) void gemm_xwt(const float* __restrict__ x,
                                                const float* __restrict__ w,
                                                float* __restrict__ y)
{
    __shared__ float red[8 * 256];   // 8 waves x 256 floats (16x16 C tile)

    const int lane  = threadIdx.x & 31;
    const int wave  = threadIdx.x >> 5;
    const int half  = lane >> 4;     // K-group within fragment
    const int mn    = lane & 15;     // A-row (batch) == B-col (output neuron)
    const int obase = blockIdx.x * 16;

    const float* xrow = x + mn * DIM;                    // A: x[m, :]
    const float* wrow = w + (size_t)(obase + mn) * DIM;  // B: w[obase+n, :]

    const int k0 = wave * 512;                           // split-K chunk
    v8f c = {};

#pragma unroll 4
    for (int k = k0; k < k0 + 512; k += 4) {
        v2f a = *(const v2f*)(xrow + k + 2 * half);           // x is L2-resident
        v2f b = __builtin_nontemporal_load(
                    (const v2f*)(wrow + k + 2 * half));       // weights stream once
        c = __builtin_amdgcn_wmma_f32_16x16x4_f32(
                /*neg_a=*/false, a, /*neg_b=*/false, b,
                /*c_mod=*/(short)0, c, /*reuse_a=*/false, /*reuse_b=*/false);
    }

    // Stage per-wave partial C tiles and reduce across the 8 K-chunks.
#pragma unroll
    for (int r = 0; r < 8; ++r)
        red[wave * 256 + r * 32 + lane] = c[r];
    __syncthreads();

    float s = 0.f;
#pragma unroll
    for (int wv = 0; wv < 8; ++wv)
        s += red[wv * 256 + threadIdx.x];

    const int r = threadIdx.x >> 5;          // C VGPR index
    const int l = threadIdx.x & 31;          // lane within layout
    const int m = r + 8 * (l >> 4);          // batch row
    const int n = l & 15;                    // output col within tile
    y[m * DIM + obase + n] = s;
}

// ---------------------------------------------------------------------------
// Flash-style decode attention, one workgroup per (b,h), 8 waves.
//   attn[b, h*128+d] = softmax(q.K^T / sqrt(128)) . V   +   v_new
//
// This kernel carries ~2GB of the 2.25GB total traffic -> it must run at
// HBM rate. Each lane owns 4 consecutive head dims, so a row is one fully
// coalesced 512B (b128/lane) non-temporal load. To maximize in-flight bytes
// per wave we process TWO rows (l, l+8) per iteration and hoist all four
// K/V row loads (2KB) ahead of the reductions. Row dot-products are reduced
// across the wave32 with a shfl_xor butterfly; per-wave flash partials
// (m, sum, acc[128]) are merged through LDS at the end.
// ---------------------------------------------------------------------------
__global__ __launch_bounds__(256) void decode_attn(const float* __restrict__ q,     // (16,4096)
                                                   const float* __restrict__ kc,    // (16,32,4096,128)
                                                   const float* __restrict__ vc,    // (16,32,4096,128)
                                                   const float* __restrict__ vnew,  // (16,4096)
                                                   float* __restrict__ attn)        // (16,4096)
{
    __shared__ float s_m[8];
    __shared__ float s_sum[8];
    __shared__ float s_acc[8][HDIM];

    const int lane = threadIdx.x & 31;
    const int wave = threadIdx.x >> 5;
    const int b    = blockIdx.x >> 5;    // 16 batches
    const int h    = blockIdx.x & 31;    // 32 heads
    const float scale = 0.08838834764831845f;   // 1/sqrt(128)

    const size_t base = ((size_t)(b * NHEADS + h)) * CACHE_LEN * HDIM;
    const float* K = kc + base + lane * 4;
    const float* V = vc + base + lane * 4;

    const v4f qv = *(const v4f*)(q + b * DIM + h * HDIM + lane * 4);

    float m   = -3.0e38f;
    float sum = 0.f;
    v4f   acc = {};

    for (int l = wave; l < CACHE_LEN; l += 16) {
        // Hoist all four streaming row loads: 2KB in flight per wave.
        v4f kv0 = __builtin_nontemporal_load((const v4f*)(K + (size_t)l       * HDIM));
        v4f kv1 = __builtin_nontemporal_load((const v4f*)(K + (size_t)(l + 8) * HDIM));
        v4f vv0 = __builtin_nontemporal_load((const v4f*)(V + (size_t)l       * HDIM));
        v4f vv1 = __builtin_nontemporal_load((const v4f*)(V + (size_t)(l + 8) * HDIM));

        float d0 = qv.x * kv0.x + qv.y * kv0.y + qv.z * kv0.z + qv.w * kv0.w;
        float d1 = qv.x * kv1.x + qv.y * kv1.y + qv.z * kv1.z + qv.w * kv1.w;
        // wave32 all-reduce (butterfly), both rows interleaved
        d0 += __shfl_xor(d0, 16, 32);  d1 += __shfl_xor(d1, 16, 32);
        d0 += __shfl_xor(d0,  8, 32);  d1 += __shfl_xor(d1,  8, 32);
        d0 += __shfl_xor(d0,  4, 32);  d1 += __shfl_xor(d1,  4, 32);
        d0 += __shfl_xor(d0,  2, 32);  d1 += __shfl_xor(d1,  2, 32);
        d0 += __shfl_xor(d0,  1, 32);  d1 += __shfl_xor(d1,  1, 32);
        const float s0 = d0 * scale;
        const float s1 = d1 * scale;

        // Online softmax update, row l
        {
            const float mn   = fmaxf(m, s0);
            const float corr = __expf(m - mn);   // first iter: exp(-3e38) == 0
            const float p    = __expf(s0 - mn);
            sum = sum * corr + p;
            acc.x = fmaf(p, vv0.x, acc.x * corr);
            acc.y = fmaf(p, vv0.y, acc.y * corr);
            acc.z = fmaf(p, vv0.z, acc.z * corr);
            acc.w = fmaf(p, vv0.w, acc.w * corr);
            m = mn;
        }
        // Online softmax update, row l+8
        {
            const float mn   = fmaxf(m, s1);
            const float corr = __expf(m - mn);
            const float p    = __expf(s1 - mn);
            sum = sum * corr + p;
            acc.x = fmaf(p, vv1.x, acc.x * corr);
            acc.y = fmaf(p, vv1.y, acc.y * corr);
            acc.z = fmaf(p, vv1.z, acc.z * corr);
            acc.w = fmaf(p, vv1.w, acc.w * corr);
            m = mn;
        }
    }

    if (lane == 0) { s_m[wave] = m; s_sum[wave] = sum; }
    *(v4f*)&s_acc[wave][lane * 4] = acc;
    __syncthreads();

    if (wave == 0) {
        float M = s_m[0];
#pragma unroll
        for (int wv = 1; wv < 8; ++wv) M = fmaxf(M, s_m[wv]);

        float tot = 0.f;
        v4f   o   = {};
#pragma unroll
        for (int wv = 0; wv < 8; ++wv) {
            const float cr = __expf(s_m[wv] - M);
            tot += s_sum[wv] * cr;
            v4f a = *(const v4f*)&s_acc[wv][lane * 4];
            o.x = fmaf(a.x, cr, o.x);
            o.y = fmaf(a.y, cr, o.y);
            o.z = fmaf(a.z, cr, o.z);
            o.w = fmaf(a.w, cr, o.w);
        }
        const float inv = 1.0f / tot;
        const v4f vn = *(const v4f*)(vnew + b * DIM + h * HDIM + lane * 4);
        v4f r;
        r.x = fmaf(o.x, inv, vn.x);   // + v_new (new-token softmax prob == 1)
        r.y = fmaf(o.y, inv, vn.y);
        r.z = fmaf(o.z, inv, vn.z);
        r.w = fmaf(o.w, inv, vn.w);
        *(v4f*)(attn + b * DIM + h * HDIM + lane * 4) = r;
    }
}

// ---------------------------------------------------------------------------
// Host launcher. Inputs: x, k_cache, v_cache, wq, wk(unused), wv, wo.
// ---------------------------------------------------------------------------
extern "C" void kernel_launch(void* const* d_in, const int* in_sizes, int n_in,
                              void* d_out, int out_size, void* d_ws, size_t ws_size,
                              hipStream_t stream)
{
    (void)in_sizes; (void)n_in; (void)out_size; (void)ws_size;

    const float* x  = (const float*)d_in[0];
    const float* kc = (const float*)d_in[1];
    const float* vc = (const float*)d_in[2];
    const float* wq = (const float*)d_in[3];
    // d_in[4] == wk: mathematically dead (single-token softmax prob == 1).
    const float* wv = (const float*)d_in[5];
    const float* wo = (const float*)d_in[6];
    float*       out = (float*)d_out;

    float* q    = (float*)d_ws;                  // (16,4096)  256 KB
    float* vnew = q    + BATCH * DIM;            // (16,4096)  256 KB
    float* attn = vnew + BATCH * DIM;            // (16,4096)  256 KB

    dim3 gblk(256);
    gemm_xwt<<<dim3(DIM / 16), gblk, 0, stream>>>(x, wq, q);
    gemm_xwt<<<dim3(DIM / 16), gblk, 0, stream>>>(x, wv, vnew);
    decode_attn<<<dim3(BATCH * NHEADS), gblk, 0, stream>>>(q, kc, vc, vnew, attn);
    gemm_xwt<<<dim3(DIM / 16), gblk, 0, stream>>>(attn, wo, out);
}